// CausalSelfAttention_85830626443607
// MI455X (gfx1250) — compile-verified
//
#include <hip/hip_runtime.h>
#include <math.h>

// Problem constants (from reference): B=4, T=1024, C=256, H=8, E=32
constexpr int Bc = 4, Tc = 1024, Cc = 256, Hc = 8, Ec = 32;

typedef float v2f __attribute__((ext_vector_type(2)));
typedef float v8f __attribute__((ext_vector_type(8)));
typedef unsigned int v4u __attribute__((ext_vector_type(4)));
typedef int v4i __attribute__((ext_vector_type(4)));
typedef int v8i __attribute__((ext_vector_type(8)));

// D = A(16x4 f32) * B(4x16 f32) + C(16x16 f32)
#define WMMA_F32(a, b, c) \
  __builtin_amdgcn_wmma_f32_16x16x4_f32(false, (a), false, (b), (short)0, (c), false, false)

// ---------------------------------------------------------------------------
// GEMM: Y[M,N] = X[M,K] @ W[N,K]^T   (nn.Linear, no bias)
// One wave per 16x16 output tile; K-loop of K/4 WMMA f32 16x16x4.
// ---------------------------------------------------------------------------
__global__ __launch_bounds__(32) void gemm_xwT(const float* __restrict__ X,
                                               const float* __restrict__ W,
                                               float* __restrict__ Y,
                                               int M, int N, int K) {
  const int lane = threadIdx.x;
  const int l16  = lane & 15;
  const int half = lane >> 4;
  const int koff = half * 2;
  const int m0 = blockIdx.y * 16;
  const int n0 = blockIdx.x * 16;

  const float* __restrict__ xrow = X + (size_t)(m0 + l16) * K;
  const float* __restrict__ wrow = W + (size_t)(n0 + l16) * K;

  v8f acc = {};
  for (int k = 0; k < K; k += 4) {
    v2f a = *(const v2f*)(xrow + k + koff);
    v2f b = *(const v2f*)(wrow + k + koff);
    acc = WMMA_F32(a, b, acc);
  }
  float* yout = Y + (size_t)(m0 + half * 8) * N + n0 + l16;
#pragma unroll
  for (int r = 0; r < 8; ++r) yout[(size_t)r * N] = acc[r];
}

// ---------------------------------------------------------------------------
// TDM: async-load one rel tile rel[q0..q0+15, k0..k0+15, h, 0..31] into LDS.
// 3D tile: dim0 = 32 contiguous floats, dim1 = 16 (j, stride H*E elems),
// dim2 = 16 (i, stride T*H*E elems).  D# layout per CDNA5 ISA ch.8.
// clang-23 toolchain exposes the 6-arg builtin:
//   (v4u group0, v8i group1, v4i group2, v4i group3, v8i unused/VADDR4, i32 cpol)
// ---------------------------------------------------------------------------
__device__ __forceinline__ void tdm_load_rel_tile(const float* rel, int q0,
                                                  int k0, int h,
                                                  unsigned lds_byte_off) {
  unsigned long long ga = (unsigned long long)(const void*)(
      rel + (((size_t)q0 * Tc + k0) * Hc + h) * Ec);
  v4u g0;
  g0.x = 1u;                                    // count=1 (valid), user mode
  g0.y = lds_byte_off;                          // LDS dest byte address
  g0.z = (unsigned)(ga & 0xffffffffu);          // global_addr[31:0]
  g0.w = (unsigned)((ga >> 32) & 0x1ffffffu)    // global_addr[56:32]
         | (2u << 30);                          // type = 2 ("image")

  v8i g1;
  g1[0] = (int)(2u << 16);                      // wg_mask=0, data_size=4B
  g1[1] = (int)(32u << 16);                     // tensor_dim0 = 32 (lo16)
  g1[2] = (int)(16u << 16);                     // dim0 hi=0 | tensor_dim1=16
  g1[3] = (int)(32u << 16);                     // dim1 hi=0 | tile_dim0 = 32
  g1[4] = (int)(16u | (16u << 16));             // tile_dim1=16, tile_dim2=16
  g1[5] = (int)(Hc * Ec);                       // tensor_dim0_stride = 256
  g1[6] = 0;                                    // stride0 hi | stride1 lo (0)
  g1[7] = (int)(((size_t)Tc * Hc * Ec) >> 16);  // stride1[47:16] (262144>>16)

  v4i g2;
  g2[0] = 16;                                   // tensor_dim2 = 16
  g2[1] = 0;                                    // tensor_dim3 = 0
  g2[2] = 0;                                    // tensor_dim2_stride lo
  g2[3] = 0;                                    // tile_dim3 = 0
  v4i g3 = {0, 0, 0, 0};                        // no dim4
  v8i g4 = {0, 0, 0, 0, 0, 0, 0, 0};            // VADDR4 slot: unused

  __builtin_amdgcn_tensor_load_to_lds(g0, g1, g2, g3, g4, 0);
}

// ---------------------------------------------------------------------------
// Flash-style causal attention with relative-position scores.
// Grid: (T/16 q-tiles, H heads); block: 1 wave.  Batch loop INSIDE so each
// rel tile (batch-independent, 1 GiB total) is streamed from HBM exactly
// once — by the Tensor Data Mover, double-buffered in LDS.
// ---------------------------------------------------------------------------
__global__ __launch_bounds__(32) void attn_kernel(const float* __restrict__ qw,
                                                  const float* __restrict__ kw,
                                                  const float* __restrict__ vw,
                                                  const float* __restrict__ rel,
                                                  float* __restrict__ yw) {
  const int qt = blockIdx.x;
  const int h  = blockIdx.y;
  const int q0 = qt * 16;
  const int lane = threadIdx.x;
  const int l16  = lane & 15;
  const int half = lane >> 4;
  const int koff = half * 2;
  const float rscale = 0.17677669529663689f;  // 1/sqrt(32)

  __shared__ float relS[2][16][16][Ec];  // TDM double buffer (64 KB)
  __shared__ float Qs[Bc][16][Ec];       // 8 KB
  __shared__ float Ks[Bc][16][Ec];       // K rows at query positions, 8 KB
  __shared__ float Sm[Bc][16][16];       // 4 KB
  __shared__ float Pm[Bc][16][16];       // 4 KB
  __shared__ float rowscale[Bc][16];
  __shared__ float mstat[Bc][16];
  __shared__ float lstat[Bc][16];

  const unsigned relS_lds0 = (unsigned)(size_t)(void*)&relS[0][0][0][0];
  const unsigned relS_lds1 = (unsigned)(size_t)(void*)&relS[1][0][0][0];

  // kick off TDM for the first k-tile
  tdm_load_rel_tile(rel, q0, 0, h, relS_lds0);

  // --- stage Q and K(query rows) tiles into LDS ---
  for (int idx = lane; idx < Bc * 16 * (Ec / 4); idx += 32) {
    int b   = idx / (16 * (Ec / 4));
    int r   = idx % (16 * (Ec / 4));
    int row = r / (Ec / 4);
    int e4  = (r % (Ec / 4)) * 4;
    size_t g = ((size_t)(b * Tc + q0 + row)) * Cc + h * Ec + e4;
    *(float4*)&Qs[b][row][e4] = *(const float4*)(qw + g);
    *(float4*)&Ks[b][row][e4] = *(const float4*)(kw + g);
  }
  if (lane < 16) {
#pragma unroll
    for (int b = 0; b < Bc; ++b) {
      mstat[b][lane] = -INFINITY;
      lstat[b][lane] = 0.0f;
    }
  }
  v8f o0[Bc], o1[Bc];
#pragma unroll
  for (int b = 0; b < Bc; ++b) { o0[b] = (v8f){}; o1[b] = (v8f){}; }
  __syncthreads();

  for (int kt = 0; kt <= qt; ++kt) {
    const int k0 = kt * 16;

    // issue TDM for the NEXT k-tile into the other buffer (overlap with S)
    if (kt < qt)
      tdm_load_rel_tile(rel, q0, k0 + 16, h, ((kt + 1) & 1) ? relS_lds1 : relS_lds0);

    // ---- S = Q K^T per batch (WMMA) ----
#pragma unroll
    for (int b = 0; b < Bc; ++b) {
      v8f s = {};
      const float* krow = kw + ((size_t)(b * Tc + k0 + l16)) * Cc + h * Ec;
      for (int kb = 0; kb < Ec; kb += 4) {
        v2f a  = *(const v2f*)&Qs[b][l16][kb + koff];
        v2f bb = *(const v2f*)(krow + kb + koff);
        s = WMMA_F32(a, bb, s);
      }
#pragma unroll
      for (int r = 0; r < 8; ++r) Sm[b][r + half * 8][l16] = s[r];
    }
    __syncthreads();

    // wait for the CURRENT rel tile (oldest outstanding TDM op)
    if (kt < qt) __builtin_amdgcn_s_wait_tensorcnt(1);
    else         __builtin_amdgcn_s_wait_tensorcnt(0);

    // ---- rel term from LDS: S[b][i][j] += rel_tile[i][j][:] . Ks[b][i][:]
    {
      const int i  = lane >> 1;
      const int j0 = (lane & 1) * 8;
      const float (*rt)[Ec] = relS[kt & 1][i];
#pragma unroll
      for (int j = 0; j < 8; ++j) {
        const float* rp = &rt[j0 + j][0];
        float4 r0 = *(const float4*)(rp + 0);
        float4 r1 = *(const float4*)(rp + 4);
        float4 r2 = *(const float4*)(rp + 8);
        float4 r3 = *(const float4*)(rp + 12);
        float4 r4 = *(const float4*)(rp + 16);
        float4 r5 = *(const float4*)(rp + 20);
        float4 r6 = *(const float4*)(rp + 24);
        float4 r7 = *(const float4*)(rp + 28);
#pragma unroll
        for (int b = 0; b < Bc; ++b) {
          const float* kq = &Ks[b][i][0];
          float s = 0.0f;
          float4 k0v = *(const float4*)(kq + 0),  k1v = *(const float4*)(kq + 4);
          float4 k2v = *(const float4*)(kq + 8),  k3v = *(const float4*)(kq + 12);
          float4 k4v = *(const float4*)(kq + 16), k5v = *(const float4*)(kq + 20);
          float4 k6v = *(const float4*)(kq + 24), k7v = *(const float4*)(kq + 28);
          s += r0.x * k0v.x + r0.y * k0v.y + r0.z * k0v.z + r0.w * k0v.w;
          s += r1.x * k1v.x + r1.y * k1v.y + r1.z * k1v.z + r1.w * k1v.w;
          s += r2.x * k2v.x + r2.y * k2v.y + r2.z * k2v.z + r2.w * k2v.w;
          s += r3.x * k3v.x + r3.y * k3v.y + r3.z * k3v.z + r3.w * k3v.w;
          s += r4.x * k4v.x + r4.y * k4v.y + r4.z * k4v.z + r4.w * k4v.w;
          s += r5.x * k5v.x + r5.y * k5v.y + r5.z * k5v.z + r5.w * k5v.w;
          s += r6.x * k6v.x + r6.y * k6v.y + r6.z * k6v.z + r6.w * k6v.w;
          s += r7.x * k7v.x + r7.y * k7v.y + r7.z * k7v.z + r7.w * k7v.w;
          Sm[b][i][j0 + j] += s;
        }
      }
    }
    __syncthreads();

    // ---- online softmax update ----
    for (int b = half; b < Bc; b += 2) {
      const int i = l16;
      float m_old = mstat[b][i];
      float mx = m_old;
      float sv[16];
#pragma unroll
      for (int j = 0; j < 16; ++j) {
        bool valid = (k0 + j) <= (q0 + i);
        float v = valid ? Sm[b][i][j] * rscale : -INFINITY;
        sv[j] = v;
        mx = fmaxf(mx, v);
      }
      float scale = __expf(m_old - mx);
      float lsum = lstat[b][i] * scale;
#pragma unroll
      for (int j = 0; j < 16; ++j) {
        float p = __expf(sv[j] - mx);
        Pm[b][i][j] = p;
        lsum += p;
      }
      mstat[b][i] = mx;
      lstat[b][i] = lsum;
      rowscale[b][i] = scale;
    }
    __syncthreads();

    // ---- O = O*rowscale + P @ V (WMMA, two column halves) ----
#pragma unroll
    for (int b = 0; b < Bc; ++b) {
#pragma unroll
      for (int r = 0; r < 8; ++r) {
        float sc = rowscale[b][r + half * 8];
        o0[b][r] *= sc;
        o1[b][r] *= sc;
      }
      for (int kb = 0; kb < 16; kb += 4) {
        v2f a = *(const v2f*)&Pm[b][l16][kb + koff];
        const int krow = k0 + kb + koff;
        const float* vb = vw + ((size_t)(b * Tc + krow)) * Cc + h * Ec + l16;
        v2f b0 = { vb[0],  vb[Cc] };
        v2f b1 = { vb[16], vb[Cc + 16] };
        o0[b] = WMMA_F32(a, b0, o0[b]);
        o1[b] = WMMA_F32(a, b1, o1[b]);
      }
    }
    __syncthreads();
  }

  // ---- normalize and write y ----
#pragma unroll
  for (int b = 0; b < Bc; ++b) {
#pragma unroll
    for (int r = 0; r < 8; ++r) {
      int row = r + half * 8;
      float inv = 1.0f / lstat[b][row];
      float* yo = yw + ((size_t)(b * Tc + q0 + row)) * Cc + h * Ec;
      yo[l16]      = o0[b][r] * inv;
      yo[l16 + 16] = o1[b][r] * inv;
    }
  }
}

// ---------------------------------------------------------------------------
extern "C" void kernel_launch(void* const* d_in, const int* in_sizes, int n_in,
                              void* d_out, int out_size, void* d_ws, size_t ws_size,
                              hipStream_t stream) {
  const float* x   = (const float*)d_in[0];
  const float* rel = (const float*)d_in[1];
  const float* Wq  = (const float*)d_in[2];
  const float* Wk  = (const float*)d_in[3];
  const float* Wv  = (const float*)d_in[4];
  const float* Wo  = (const float*)d_in[5];
  float* out = (float*)d_out;

  const size_t NTC = (size_t)Bc * Tc * Cc;  // 4 MB each
  float* qw = (float*)d_ws;
  float* kw = qw + NTC;
  float* vw = kw + NTC;
  float* yw = vw + NTC;

  dim3 gg(Cc / 16, (Bc * Tc) / 16);
  gemm_xwT<<<gg, 32, 0, stream>>>(x, Wq, qw, Bc * Tc, Cc, Cc);
  gemm_xwT<<<gg, 32, 0, stream>>>(x, Wk, kw, Bc * Tc, Cc, Cc);
  gemm_xwT<<<gg, 32, 0, stream>>>(x, Wv, vw, Bc * Tc, Cc, Cc);

  attn_kernel<<<dim3(Tc / 16, Hc), 32, 0, stream>>>(qw, kw, vw, rel, yw);

  gemm_xwT<<<gg, 32, 0, stream>>>(yw, Wo, out, Bc * Tc, Cc, Cc);
}